// HGSRModel_77799037600107
// MI455X (gfx1250) — compile-verified
//
#include <hip/hip_runtime.h>
#include <cstdint>
#include <cstddef>

// ---------------------------------------------------------------------------
// HGSR hyperbolic GNN on MI455X (gfx1250)
//  - wave32 shuffle reductions for row norms (logmap0 / expmap0 / proj)
//  - SpMM scatter via GLOBAL_LOAD_ASYNC_TO_LDS_B128 (512B row per instr),
//    double-buffered: chunk k+1 DMA overlaps chunk k atomic scatter,
//    synchronized with s_wait_asynccnt <= CHUNK (async loads retire in order)
//  - hw f32 global atomics (global_atomic_add_f32) for the scatter
//  - cur/next buffers (76.8 MB each) are L2-resident (192 MB L2)
// ---------------------------------------------------------------------------

#define N_NODES   150000
#define NUM_USERS 50000
#define DIM       64
#define D2        128
#define E_UV      1200000
#define E_UU      600000
#define F_EPS     1e-7f
#define F_MINNORM 1e-15f
#define CHUNK     4        // edges per pipeline stage
#define PIPE_ITER 4        // target chunks per wave (grid sizing)

typedef int v4i __attribute__((vector_size(16)));
typedef __attribute__((address_space(1))) v4i   as1_v4i;
typedef __attribute__((address_space(3))) v4i   as3_v4i;
typedef __attribute__((address_space(3))) float as3_float;

__device__ __forceinline__ float waveReduceSum(float v) {
#pragma unroll
  for (int m = 16; m >= 1; m >>= 1) v += __shfl_xor(v, m, 32);
  return v;
}

// One async DMA: 32 lanes x 16B = 512B (one 128-float row) global -> LDS.
__device__ __forceinline__ void async_gather_b128(const float* gsrc, float* lds_dst) {
#if __has_builtin(__builtin_amdgcn_global_load_async_to_lds_b128)
  __builtin_amdgcn_global_load_async_to_lds_b128(
      (as1_v4i*)gsrc, (as3_v4i*)lds_dst, 0, 0);
#else
  unsigned l = (unsigned)(unsigned long long)(as3_float*)lds_dst;
  unsigned long long g = (unsigned long long)(const void*)gsrc;
  asm volatile("global_load_async_to_lds_b128 %0, %1, off"
               :: "v"(l), "v"(g) : "memory");
#endif
}

template <int N>
__device__ __forceinline__ void wait_asynccnt() {
#if __has_builtin(__builtin_amdgcn_s_wait_asynccnt)
  __builtin_amdgcn_s_wait_asynccnt(N);
#else
  asm volatile("s_wait_asynccnt %0" :: "i"(N) : "memory");
#endif
}

__device__ __forceinline__ void atomAddF32(float* p, float v) {
  __hip_atomic_fetch_add(p, v, __ATOMIC_RELAXED, __HIP_MEMORY_SCOPE_AGENT);
}

// ---------------------------------------------------------------------------
// Kernel 1: cur[i] = [logmap0(proj(emb_i)), (i<U ? logmap0(proj(usf_i))
//                                                 : logmap0(proj(emb_i)))]
// One wave per node; lane handles elements 2l, 2l+1 of the 64-dim row.
// K = 1 (C = 1), sqrtK = 1.
// ---------------------------------------------------------------------------
__device__ __forceinline__ float2 logmap_row(float2 xv, int lane) {
  // y = row[1:]; lane 0's element 0 is excluded from y
  float contrib = xv.y * xv.y + ((lane == 0) ? 0.0f : xv.x * xv.x);
  float ysq   = waveReduceSum(contrib);
  float ynorm = fmaxf(sqrtf(ysq), F_MINNORM);
  float x0    = sqrtf(fmaxf(1.0f + ysq, F_EPS));        // proj
  float th    = fmaxf(x0, 1.0f + F_EPS);                // clip
  float arc   = logf(th + sqrtf(fmaxf(th * th - 1.0f, 0.0f)));
  float s     = arc / ynorm;
  float2 t;
  t.x = (lane == 0) ? 0.0f : s * xv.x;
  t.y = s * xv.y;
  return t;
}

__global__ __launch_bounds__(256)
void build_cur_kernel(const float* __restrict__ emb,
                      const float* __restrict__ usf,
                      float* __restrict__ cur) {
  const int wave = blockIdx.x * (blockDim.x >> 5) + (threadIdx.x >> 5);
  const int lane = threadIdx.x & 31;
  if (wave >= N_NODES) return;

  const float2 xv = ((const float2*)(emb + (size_t)wave * DIM))[lane];
  float2 t = logmap_row(xv, lane);

  float2* outrow = (float2*)(cur + (size_t)wave * D2);
  outrow[lane] = t;                       // first 64 dims: x_tan

  float2 t2;
  if (wave < NUM_USERS) {                 // wave-uniform branch
    const float2 uv2 = ((const float2*)(usf + (size_t)wave * DIM))[lane];
    t2 = logmap_row(uv2, lane);
  } else {
    t2 = t;                               // z_tan = x_tan for item nodes
  }
  outrow[32 + lane] = t2;                 // second 64 dims: z_tan
}

// ---------------------------------------------------------------------------
// Kernel 2: zero fill (float4 grid-stride)
// ---------------------------------------------------------------------------
__global__ __launch_bounds__(256)
void zero_kernel(float4* __restrict__ p, int n4) {
  const float4 z = make_float4(0.f, 0.f, 0.f, 0.f);
  for (int i = blockIdx.x * blockDim.x + threadIdx.x; i < n4;
       i += gridDim.x * blockDim.x)
    p[i] = z;
}

// ---------------------------------------------------------------------------
// Kernel 3: SpMM scatter  dst[row] += w*val * src[col]  (128 floats/edge)
// Double-buffered async pipeline per wave:
//   issue chunk k+1 -> LDS slot (k+1)&1 ; s_wait_asynccnt<=CHUNK
//   (chunk k resident, k+1 in flight)   ; scatter chunk k with f32 atomics.
// ---------------------------------------------------------------------------
__device__ __forceinline__ void issue_chunk(const int* __restrict__ cols, int nE,
                                            const float* __restrict__ src,
                                            int lane, long long base,
                                            float* ldsSlot) {
#pragma unroll
  for (int j = 0; j < CHUNK; ++j) {
    long long e = base + j;
    int c = (e < nE) ? cols[e] : 0;            // clamp tail: harmless load
    async_gather_b128(src + (size_t)c * D2 + lane * 4,
                      ldsSlot + j * D2 + lane * 4);
  }
}

__device__ __forceinline__ void process_chunk(const int* __restrict__ rows,
                                              const float* __restrict__ vals,
                                              int nE, float w,
                                              float* __restrict__ dst,
                                              int lane, long long base,
                                              const float* ldsSlot) {
#pragma unroll
  for (int j = 0; j < CHUNK; ++j) {
    long long e = base + j;
    if (e < nE) {
      const float sv = w * vals[e];
      const int   r  = rows[e];
      const float4 v = *(const float4*)(ldsSlot + j * D2 + lane * 4);
      float* dp = dst + (size_t)r * D2 + lane * 4;
      atomAddF32(dp + 0, sv * v.x);
      atomAddF32(dp + 1, sv * v.y);
      atomAddF32(dp + 2, sv * v.z);
      atomAddF32(dp + 3, sv * v.w);
    }
  }
}

__global__ __launch_bounds__(256)
void spmm_scatter_kernel(const int*   __restrict__ rows,
                         const int*   __restrict__ cols,
                         const float* __restrict__ vals,
                         int nEdges, float w,
                         const float* __restrict__ src,
                         float*       __restrict__ dst) {
  __shared__ float shm[8 * 2 * CHUNK * D2];  // 8 waves * 2 bufs * 4 rows = 32 KB
  const int lane       = threadIdx.x & 31;
  const int waveInBlk  = threadIdx.x >> 5;
  const int wavesPerBl = blockDim.x >> 5;
  const long long waveId = (long long)blockIdx.x * wavesPerBl + waveInBlk;
  const long long nWaves = (long long)gridDim.x * wavesPerBl;
  const long long step   = nWaves * CHUNK;
  float* lds0 = &shm[waveInBlk * 2 * CHUNK * D2];

  const long long base0 = waveId * CHUNK;
  if (base0 >= nEdges) return;               // wave-uniform

  issue_chunk(cols, nEdges, src, lane, base0, lds0);   // prologue: chunk 0
  int sel = 0;
  for (long long base = base0; base < nEdges; base += step) {
    const long long next = base + step;
    if (next < nEdges) {
      issue_chunk(cols, nEdges, src, lane, next, lds0 + (sel ^ 1) * CHUNK * D2);
      wait_asynccnt<CHUNK>();  // chunk `base` done (in-order), `next` in flight
    } else {
      wait_asynccnt<0>();      // epilogue: drain
    }
    process_chunk(rows, vals, nEdges, w, dst, lane, base,
                  lds0 + sel * CHUNK * D2);
    sel ^= 1;
  }
}

// ---------------------------------------------------------------------------
// Kernel 4: acc += buf (float4 grid-stride)
// ---------------------------------------------------------------------------
__global__ __launch_bounds__(256)
void accum_kernel(float4* __restrict__ acc, const float4* __restrict__ add, int n4) {
  for (int i = blockIdx.x * blockDim.x + threadIdx.x; i < n4;
       i += gridDim.x * blockDim.x) {
    float4 a = acc[i], b = add[i];
    a.x += b.x; a.y += b.y; a.z += b.z; a.w += b.w;
    acc[i] = a;
  }
}

// ---------------------------------------------------------------------------
// Kernel 5: out = proj(expmap0(acc))  in place on d_out (one wave per node;
// lane handles elements 4l..4l+3 of the 128-dim row).
// expmap0: x = acc[1:], xn = max(||x||,1e-15), rest = sinh(xn)/xn * x;
// proj:    out0 = sqrt(max(1 + ||rest||^2, EPS)), out[1:] = rest.
// ---------------------------------------------------------------------------
__global__ __launch_bounds__(256)
void finalize_kernel(float* __restrict__ acc) {
  const int wave = blockIdx.x * (blockDim.x >> 5) + (threadIdx.x >> 5);
  const int lane = threadIdx.x & 31;
  if (wave >= N_NODES) return;

  float4* rowp = (float4*)(acc + (size_t)wave * D2);
  float4 v = rowp[lane];
  float contrib = v.y * v.y + v.z * v.z + v.w * v.w +
                  ((lane == 0) ? 0.0f : v.x * v.x);
  float xsq = waveReduceSum(contrib);
  float xn  = fmaxf(sqrtf(xsq), F_MINNORM);
  float e   = expf(xn);
  float ei  = 1.0f / e;
  float sh  = 0.5f * (e - ei);
  float s   = sh / xn;                 // sinh(theta)/x_norm
  float restsq = s * s * xsq;          // ||rest||^2
  float out0   = sqrtf(fmaxf(1.0f + restsq, F_EPS));
  float4 o;
  o.x = (lane == 0) ? out0 : s * v.x;
  o.y = s * v.y;
  o.z = s * v.z;
  o.w = s * v.w;
  rowp[lane] = o;
}

// ---------------------------------------------------------------------------
extern "C" void kernel_launch(void* const* d_in, const int* in_sizes, int n_in,
                              void* d_out, int out_size, void* d_ws, size_t ws_size,
                              hipStream_t stream) {
  const float* emb    = (const float*)d_in[0];
  const float* usf    = (const float*)d_in[1];
  const int*   uv_row = (const int*)d_in[2];
  const int*   uv_col = (const int*)d_in[3];
  const float* uv_val = (const float*)d_in[4];
  const int*   uu_row = (const int*)d_in[5];
  const int*   uu_col = (const int*)d_in[6];
  const float* uu_val = (const float*)d_in[7];
  float* out = (float*)d_out;

  const size_t rowElems = (size_t)N_NODES * D2;     // 19.2M floats (76.8 MB)
  float* bufA = (float*)d_ws;                       // cur  (layer input)
  float* bufB = bufA + rowElems;                    // next (layer output)
  const int n4 = (int)(rowElems / 4);

  const int WAVE_BLOCKS = N_NODES / 8;                                  // 18750
  // ~PIPE_ITER chunks per wave so the async double-buffer pipeline fills
  const int UV_BLOCKS = (E_UV + CHUNK * 8 * PIPE_ITER - 1) / (CHUNK * 8 * PIPE_ITER); // 9375
  const int UU_BLOCKS = (E_UU + CHUNK * 8 * PIPE_ITER - 1) / (CHUNK * 8 * PIPE_ITER); // 4688
  const int ELT_BLOCKS = 2048;

  // Stage 1: tangent features
  build_cur_kernel<<<WAVE_BLOCKS, 256, 0, stream>>>(emb, usf, bufA);
  zero_kernel<<<ELT_BLOCKS, 256, 0, stream>>>((float4*)out,  n4);  // acc = 0
  zero_kernel<<<ELT_BLOCKS, 256, 0, stream>>>((float4*)bufB, n4);

  // Layer 1: bufB = 0.7*A_uv@bufA + 0.3*A_uu@bufA ; acc += bufB
  spmm_scatter_kernel<<<UV_BLOCKS, 256, 0, stream>>>(uv_row, uv_col, uv_val,
                                                     E_UV, 0.7f, bufA, bufB);
  spmm_scatter_kernel<<<UU_BLOCKS, 256, 0, stream>>>(uu_row, uu_col, uu_val,
                                                     E_UU, 0.3f, bufA, bufB);
  accum_kernel<<<ELT_BLOCKS, 256, 0, stream>>>((float4*)out, (const float4*)bufB, n4);

  // Layer 2: bufA = 0.7*A_uv@bufB + 0.3*A_uu@bufB ; acc += bufA
  zero_kernel<<<ELT_BLOCKS, 256, 0, stream>>>((float4*)bufA, n4);
  spmm_scatter_kernel<<<UV_BLOCKS, 256, 0, stream>>>(uv_row, uv_col, uv_val,
                                                     E_UV, 0.7f, bufB, bufA);
  spmm_scatter_kernel<<<UU_BLOCKS, 256, 0, stream>>>(uu_row, uu_col, uu_val,
                                                     E_UU, 0.3f, bufB, bufA);
  accum_kernel<<<ELT_BLOCKS, 256, 0, stream>>>((float4*)out, (const float4*)bufA, n4);

  // Stage 3: expmap0 + proj, in place on d_out
  finalize_kernel<<<WAVE_BLOCKS, 256, 0, stream>>>(out);
}